// SAGELayer_10453950399133
// MI455X (gfx1250) — compile-verified
//
#include <hip/hip_runtime.h>

#define N_NODES 50000
#define N_EDGES 600000
#define D_IN 128
#define D_OUT 256
#define R_TILES 5               // row tiles (of 16) per wave in the GEMM

typedef float v2f __attribute__((ext_vector_type(2)));
typedef float v8f __attribute__((ext_vector_type(8)));

// ---------------------------------------------------------------------------
// Kernel 0: zero the aggregation buffer x[N_NODES][D_IN] in workspace.
// 6,400,000 floats = 1,600,000 float4 stores; exact grid, no tail.
// ---------------------------------------------------------------------------
__global__ __launch_bounds__(256) void sage_zero_x(float4* __restrict__ x4) {
  int i = blockIdx.x * 256 + threadIdx.x;
  x4[i] = make_float4(0.f, 0.f, 0.f, 0.f);
}

// ---------------------------------------------------------------------------
// Kernel 1: transpose W [128][256] -> WT [256][128] (128 KB, one-shot).
// Makes the GEMM B-operand pair {W[k][n], W[k+1][n]} contiguous in memory so
// it lowers to a single global_load_b64 per K-step.
// ---------------------------------------------------------------------------
__global__ __launch_bounds__(256) void sage_transpose_w(
    const float* __restrict__ W, float* __restrict__ WT) {
  int idx = blockIdx.x * 256 + threadIdx.x;   // 0 .. 32767
  int n = idx >> 7;                           // 0 .. 255
  int k = idx & 127;                          // 0 .. 127
  WT[idx] = W[(size_t)k * D_OUT + n];
}

// ---------------------------------------------------------------------------
// Kernel 2: edge-parallel weighted scatter-add (SpMM).
// One wave32 per edge; each lane handles 4 consecutive features (32*4 = 128).
// h (25.6 MB) and x (25.6 MB) are L2-resident (192 MB L2), so both the random
// gather and the fp32 atomics run at L2 speed. unsafeAtomicAdd lowers to
// global_atomic_add_f32 (confirmed: no CAS loop in round-1 histogram).
// ---------------------------------------------------------------------------
__global__ __launch_bounds__(256) void sage_aggregate(
    const float* __restrict__ h,
    const int* __restrict__ edge_src,
    const int* __restrict__ edge_dst,
    const float* __restrict__ edge_weight,
    float* __restrict__ x) {
  const int wave = (blockIdx.x * 256 + threadIdx.x) >> 5;  // edge id
  const int lane = threadIdx.x & 31;

  const int   s = edge_src[wave];
  const int   d = edge_dst[wave];
  const float w = edge_weight[wave];

  const float4 v = *(const float4*)(h + (size_t)s * D_IN + lane * 4);
  float* xp = x + (size_t)d * D_IN + lane * 4;

  unsafeAtomicAdd(xp + 0, w * v.x);
  unsafeAtomicAdd(xp + 1, w * v.y);
  unsafeAtomicAdd(xp + 2, w * v.z);
  unsafeAtomicAdd(xp + 3, w * v.w);
}

// ---------------------------------------------------------------------------
// Kernel 3: out = relu(x @ W) with V_WMMA_F32_16X16X4_F32 (full fp32).
// One wave computes an 80x16 output strip: R_TILES=5 row tiles sharing one
// B (column) operand, so per K-step = 5 A-loads(b64) + 1 B-load(b64) +
// 5 WMMAs -> 1.2 loads per WMMA (was 3.0 in round 1).
//
// VGPR layouts (ISA 7.12.2):
//   A 16x4 f32 : lanes 0-15 -> M=lane, {K=k,k+1}; lanes 16-31 -> {K=k+2,k+3}
//   B 4x16 f32 : lanes 0-15 -> N=lane, {K=k,k+1}; lanes 16-31 -> {K=k+2,k+3}
//     (loaded from WT[n][k..k+1], contiguous b64)
//   C/D 16x16  : VGPR i = row (m0+i) lanes 0-15 / row (m0+8+i) lanes 16-31
// ---------------------------------------------------------------------------
__global__ __launch_bounds__(256) void sage_gemm_relu(
    const float* __restrict__ x,    // [N_NODES][D_IN]
    const float* __restrict__ WT,   // [D_OUT][D_IN]  (transposed weights)
    float* __restrict__ out) {      // [N_NODES][D_OUT]
  const int waveId = (blockIdx.x * 256 + threadIdx.x) >> 5;
  const int lane   = threadIdx.x & 31;

  const int group   = waveId >> 4;          // 0..624  (row-tile group of 5)
  const int colTile = waveId & 15;          // 0..15
  const int m0 = group * (R_TILES * 16);
  const int n0 = colTile * 16;

  const int laneHalf = lane >> 4;           // 0: K pair {k,k+1}, 1: {k+2,k+3}
  const int laneIdx  = lane & 15;           // M for A, N for B

  const float* aBase = x  + (size_t)(m0 + laneIdx) * D_IN + laneHalf * 2;
  const float* bBase = WT + (size_t)(n0 + laneIdx) * D_IN + laneHalf * 2;

  v8f c[R_TILES];
#pragma unroll
  for (int r = 0; r < R_TILES; ++r) c[r] = (v8f){};

#pragma unroll
  for (int k = 0; k < D_IN; k += 4) {
    const v2f b = *(const v2f*)(bBase + k);
#pragma unroll
    for (int r = 0; r < R_TILES; ++r) {
      const v2f a = *(const v2f*)(aBase + (size_t)r * 16 * D_IN + k);
      c[r] = __builtin_amdgcn_wmma_f32_16x16x4_f32(
          /*neg_a=*/false, a, /*neg_b=*/false, b,
          /*c_mod=*/(short)0, c[r], /*reuse_a=*/false, /*reuse_b=*/false);
    }
  }

  // ReLU + store: accum VGPR i -> row (m0 + r*16 + laneHalf*8 + i), col n0+laneIdx
#pragma unroll
  for (int r = 0; r < R_TILES; ++r) {
    float* o = out + (size_t)(m0 + r * 16 + laneHalf * 8) * D_OUT + n0 + laneIdx;
#pragma unroll
    for (int i = 0; i < 8; ++i) {
      o[(size_t)i * D_OUT] = fmaxf(c[r][i], 0.0f);
    }
  }
}

// ---------------------------------------------------------------------------
extern "C" void kernel_launch(void* const* d_in, const int* in_sizes, int n_in,
                              void* d_out, int out_size, void* d_ws, size_t ws_size,
                              hipStream_t stream) {
  (void)in_sizes; (void)n_in; (void)out_size; (void)ws_size;

  const float* h    = (const float*)d_in[0];   // [50000,128] f32
  const int*   esrc = (const int*)  d_in[1];   // [600000] i32
  const int*   edst = (const int*)  d_in[2];   // [600000] i32
  const float* ew   = (const float*)d_in[3];   // [600000] f32
  const float* Wn   = (const float*)d_in[4];   // [128,256] f32
  float*       out  = (float*)d_out;           // [50000,256] f32

  float* x  = (float*)d_ws;                    // 50000*128 f32 = 25.6 MB
  float* WT = x + (size_t)N_NODES * D_IN;      // + 256*128 f32 = 128 KB

  // 1) zero x               (6250 blocks * 256 threads * float4 = 6.4M floats)
  sage_zero_x<<<(N_NODES * D_IN / 4) / 256, 256, 0, stream>>>((float4*)x);

  // 2) transpose W -> WT    (32768 elements)
  sage_transpose_w<<<(D_IN * D_OUT) / 256, 256, 0, stream>>>(Wn, WT);

  // 3) weighted scatter-add (one wave per edge: 600000 waves / 8 per block)
  sage_aggregate<<<N_EDGES / 8, 256, 0, stream>>>(h, esrc, edst, ew, x);

  // 4) fp32 WMMA GEMM+ReLU  (625 groups * 16 col tiles = 10000 waves)
  sage_gemm_relu<<<(N_NODES / (16 * R_TILES)) * (D_OUT / 16) / 8, 256, 0, stream>>>(
      x, WT, out);
}